// ann1_snn7_57208964383543
// MI455X (gfx1250) — compile-verified
//
#include <hip/hip_runtime.h>
#include <math.h>

// ---------------- problem constants ----------------
#define BATCH 32
#define TSTEPS 25

constexpr float ETA_M  = 0.778800783071404870f;   // e^-1/4
constexpr float AX_A1  = 1.146680224242847190f;   // e^-1/4 + e^-1
constexpr float AX_A2  = -0.286504796860190100f;  // -e^-5/4
constexpr float V_TH   = 1.0f;

typedef __attribute__((ext_vector_type(2))) float v2f;
typedef __attribute__((ext_vector_type(8))) float v8f;

__device__ __forceinline__ v8f wmma_f32_16x16x4(v2f a, v2f b, v8f c) {
  // D = A(16x4) * B(4x16) + C, f32, wave32
  return __builtin_amdgcn_wmma_f32_16x16x4_f32(false, a, false, b, (short)0, c, false, false);
}

// ---------------- utility kernels ----------------
__global__ void zero_kernel(float* __restrict__ p, size_t n) {
  size_t i = (size_t)blockIdx.x * blockDim.x + threadIdx.x;
  size_t stride = (size_t)gridDim.x * blockDim.x;
  for (; i < n; i += stride) p[i] = 0.0f;
}

// ANN front-end: conv 3->32, 3x3 VALID on 32x32 -> 30x30, + bias, sigmoid. Once per launch.
__global__ void ann1_kernel(const float* __restrict__ in, const float* __restrict__ w,
                            const float* __restrict__ bias, float* __restrict__ frame) {
  const int n = BATCH * 32 * 30 * 30;
  int i = blockIdx.x * blockDim.x + threadIdx.x;
  if (i >= n) return;
  int ow = i % 30;
  int oh = (i / 30) % 30;
  int co = (i / 900) % 32;
  int b  = i / (900 * 32);
  float acc = bias[co];
  #pragma unroll
  for (int ci = 0; ci < 3; ++ci)
    #pragma unroll
    for (int kh = 0; kh < 3; ++kh)
      #pragma unroll
      for (int kw = 0; kw < 3; ++kw)
        acc += in[(((size_t)b * 3 + ci) * 32 + (oh + kh)) * 32 + (ow + kw)] *
               w[(((size_t)co * 3 + ci) * 3 + kh) * 3 + kw];
  frame[i] = 1.0f / (1.0f + __expf(-acc));
}

// Standalone axon IIR (ax2 on the constant frame): y = a1*y1 + a2*y2 + x
__global__ void axon_kernel(const float* __restrict__ x, float* __restrict__ y1,
                            float* __restrict__ y2, float* __restrict__ out, int n) {
  int i = blockIdx.x * blockDim.x + threadIdx.x;
  if (i >= n) return;
  float a = y1[i], b = y2[i];
  float y = AX_A1 * a + AX_A2 * b + x[i];
  out[i] = y;
  y2[i] = a;
  y1[i] = y;
}

// 2x2 maxpool (stride 2, VALID) fused with the next layer's axon IIR.
__global__ void maxpool_axon_kernel(const float* __restrict__ in,
                                    float* __restrict__ ay1, float* __restrict__ ay2,
                                    float* __restrict__ aout,
                                    int C, int Hin, int Win, int Hout, int Wout) {
  const int n = BATCH * C * Hout * Wout;
  int i = blockIdx.x * blockDim.x + threadIdx.x;
  if (i >= n) return;
  int w = i % Wout;
  int h = (i / Wout) % Hout;
  int c = (i / (Wout * Hout)) % C;
  int b = i / (Wout * Hout * C);
  const float* p = in + (((size_t)b * C + c) * Hin + 2 * h) * Win + 2 * w;
  float m = fmaxf(fmaxf(p[0], p[1]), fmaxf(p[Win], p[Win + 1]));
  float y1v = ay1[i], y2v = ay2[i];
  float y = AX_A1 * y1v + AX_A2 * y2v + m;
  aout[i] = y;
  ay2[i] = y1v;
  ay1[i] = y;
}

// Implicit-GEMM 3x3 VALID conv as 9 shifted GEMMs (K = Cin per tap), fused with
// LIF update and the next layer's axon IIR. Each wave computes a 16x32 output
// tile (two accumulators share every A fetch). Requires Cout % 32 == 0 and
// BATCH*Hout*Wout % 16 == 0 (true for all layers here). Pure pointer-stride
// addressing in the K loop: no divisions per WMMA.
__global__ void conv_wmma_lif_axon(const float* __restrict__ in,   // [B,Cin,Hin,Win]
                                   const float* __restrict__ w,    // [Cout,Cin,3,3]
                                   const float* __restrict__ bias, // [Cout]
                                   float* __restrict__ vstate,     // [B,Cout,Hout,Wout]
                                   float* __restrict__ sstate,     // spike state / output
                                   float* __restrict__ ay1, float* __restrict__ ay2,
                                   float* __restrict__ aout,       // axon output for next layer
                                   int Cin, int Cout, int Hin, int Win,
                                   int Hout, int Wout, int totalTiles) {
  const int lane = threadIdx.x & 31;
  const int waveInBlock = threadIdx.x >> 5;
  const int tileId = blockIdx.x * (blockDim.x >> 5) + waveInBlock;
  if (tileId >= totalTiles) return;

  const int nPairs = Cout >> 5;               // 16x32 tiles along N
  const int mTile = tileId / nPairs;
  const int nPair = tileId % nPairs;
  const int l16 = lane & 15;
  const int lh  = lane >> 4;                  // which K-half this lane feeds

  // A-matrix row for this lane (both lane halves cover M rows 0..15 of the tile)
  const int mRow = mTile * 16 + l16;
  const int hw   = Hout * Wout;
  const int b    = mRow / hw;
  const int r    = mRow % hw;
  const int oh   = r / Wout;
  const int ow   = r % Wout;
  const int hwin = Hin * Win;
  const float* inB = in + (size_t)b * Cin * hwin;

  const int col0 = nPair * 32 + l16;          // B-matrix columns for this lane
  const int col1 = col0 + 16;

  v8f acc0 = {0.f, 0.f, 0.f, 0.f, 0.f, 0.f, 0.f, 0.f};
  v8f acc1 = {0.f, 0.f, 0.f, 0.f, 0.f, 0.f, 0.f, 0.f};

  #pragma unroll
  for (int kh = 0; kh < 3; ++kh) {
    #pragma unroll
    for (int kw = 0; kw < 3; ++kw) {
      const int tap = kh * 3 + kw;
      const float* ap  = inB + (oh + kh) * Win + (ow + kw) + (size_t)(lh * 2) * hwin;
      const float* bp0 = w + (size_t)col0 * Cin * 9 + tap + (size_t)(lh * 2) * 9;
      const float* bp1 = w + (size_t)col1 * Cin * 9 + tap + (size_t)(lh * 2) * 9;
      for (int c = 0; c < Cin; c += 4) {
        v2f av;  av.x  = ap[0];   av.y  = ap[hwin];
        v2f bv0; bv0.x = bp0[0];  bv0.y = bp0[9];
        v2f bv1; bv1.x = bp1[0];  bv1.y = bp1[9];
        acc0 = wmma_f32_16x16x4(av, bv0, acc0);
        acc1 = wmma_f32_16x16x4(av, bv1, acc1);
        ap  += 4 * (size_t)hwin;
        bp0 += 36;
        bp1 += 36;
      }
    }
  }

  // Epilogue: bias + LIF + next-layer axon on both 16x16 tiles.
  // D layout: VGPR g holds M = mTile*16 + g + 8*lh, N = col (l16 within tile).
  const float bval0 = bias[col0];
  const float bval1 = bias[col1];
  int mrBase = mTile * 16 + 8 * lh;
  int bb  = mrBase / hw;
  int pix = mrBase % hw;
  #pragma unroll
  for (int g = 0; g < 8; ++g) {
    const size_t rowBase = (size_t)bb * Cout * hw + pix;
    {
      size_t oidx = rowBase + (size_t)col0 * hw;
      float cur  = acc0[g] + bval0;
      float vold = vstate[oidx], sold = sstate[oidx];
      float vnew = ETA_M * vold * (1.0f - sold) + cur;
      float spk  = (vnew > V_TH) ? 1.0f : 0.0f;
      vstate[oidx] = vnew;
      sstate[oidx] = spk;
      float y1v = ay1[oidx], y2v = ay2[oidx];
      float y = AX_A1 * y1v + AX_A2 * y2v + spk;
      aout[oidx] = y;
      ay2[oidx] = y1v;
      ay1[oidx] = y;
    }
    {
      size_t oidx = rowBase + (size_t)col1 * hw;
      float cur  = acc1[g] + bval1;
      float vold = vstate[oidx], sold = sstate[oidx];
      float vnew = ETA_M * vold * (1.0f - sold) + cur;
      float spk  = (vnew > V_TH) ? 1.0f : 0.0f;
      vstate[oidx] = vnew;
      sstate[oidx] = spk;
      float y1v = ay1[oidx], y2v = ay2[oidx];
      float y = AX_A1 * y1v + AX_A2 * y2v + spk;
      aout[oidx] = y;
      ay2[oidx] = y1v;
      ay1[oidx] = y;
    }
    if (++pix == hw) { pix = 0; ++bb; }   // next M row, carry into batch
  }
}

// Fully-connected layer via WMMA + LIF; each wave computes a 16x32 tile
// (two N-tiles per A fetch). Optionally fused axon on output (fc7) or final
// spike write to d_out [B,10,T] (fc8; columns masked against Nvalid).
__global__ void fc_wmma_lif(const float* __restrict__ in,  // [B,K]
                            const float* __restrict__ w,   // [Nvalid,K]
                            const float* __restrict__ bias,
                            float* __restrict__ vstate, float* __restrict__ sstate,
                            float* __restrict__ ay1, float* __restrict__ ay2,
                            float* __restrict__ aout,      // may be null (fc8)
                            float* __restrict__ outT,      // may be null (fc7); [B,10,T]
                            int K, int nPairs, int Nvalid, int t) {
  const int lane = threadIdx.x & 31;
  const int wave = (blockIdx.x * blockDim.x + threadIdx.x) >> 5;
  const int totalTiles = 2 * nPairs;          // M = BATCH = 32 -> 2 M-tiles
  if (wave >= totalTiles) return;
  const int mTile = wave / nPairs;
  const int nPair = wave % nPairs;
  const int l16 = lane & 15;
  const int lh  = lane >> 4;

  const int row  = mTile * 16 + l16;          // batch index (A rows)
  const int col0 = nPair * 32 + l16;          // output neurons (B cols / D cols)
  const int col1 = col0 + 16;
  const bool ok0 = (col0 < Nvalid);
  const bool ok1 = (col1 < Nvalid);
  const float* ap  = in + (size_t)row * K + lh * 2;
  const float* bp0 = ok0 ? (w + (size_t)col0 * K + lh * 2) : nullptr;
  const float* bp1 = ok1 ? (w + (size_t)col1 * K + lh * 2) : nullptr;

  v8f acc0 = {0.f, 0.f, 0.f, 0.f, 0.f, 0.f, 0.f, 0.f};
  v8f acc1 = {0.f, 0.f, 0.f, 0.f, 0.f, 0.f, 0.f, 0.f};
  for (int kk = 0; kk < K; kk += 4) {
    v2f av;  av.x = ap[0]; av.y = ap[1];
    v2f bv0 = {0.f, 0.f};
    v2f bv1 = {0.f, 0.f};
    if (ok0) { bv0.x = bp0[0]; bv0.y = bp0[1]; bp0 += 4; }
    if (ok1) { bv1.x = bp1[0]; bv1.y = bp1[1]; bp1 += 4; }
    acc0 = wmma_f32_16x16x4(av, bv0, acc0);
    acc1 = wmma_f32_16x16x4(av, bv1, acc1);
    ap += 4;
  }

  const float bval0 = ok0 ? bias[col0] : 0.f;
  const float bval1 = ok1 ? bias[col1] : 0.f;
  #pragma unroll
  for (int g = 0; g < 8; ++g) {
    const int bo = mTile * 16 + g + 8 * lh;   // batch row of this D element
    if (ok0) {
      size_t idx = (size_t)bo * Nvalid + col0;
      float cur  = acc0[g] + bval0;
      float vold = vstate[idx], sold = sstate[idx];
      float vnew = ETA_M * vold * (1.0f - sold) + cur;
      float spk  = (vnew > V_TH) ? 1.0f : 0.0f;
      vstate[idx] = vnew;
      sstate[idx] = spk;
      if (aout) {
        float y1v = ay1[idx], y2v = ay2[idx];
        float y = AX_A1 * y1v + AX_A2 * y2v + spk;
        aout[idx] = y;
        ay2[idx] = y1v;
        ay1[idx] = y;
      }
      if (outT) outT[((size_t)bo * 10 + col0) * TSTEPS + t] = spk;
    }
    if (ok1) {
      size_t idx = (size_t)bo * Nvalid + col1;
      float cur  = acc1[g] + bval1;
      float vold = vstate[idx], sold = sstate[idx];
      float vnew = ETA_M * vold * (1.0f - sold) + cur;
      float spk  = (vnew > V_TH) ? 1.0f : 0.0f;
      vstate[idx] = vnew;
      sstate[idx] = spk;
      if (aout) {
        float y1v = ay1[idx], y2v = ay2[idx];
        float y = AX_A1 * y1v + AX_A2 * y2v + spk;
        aout[idx] = y;
        ay2[idx] = y1v;
        ay1[idx] = y;
      }
      if (outT) outT[((size_t)bo * 10 + col1) * TSTEPS + t] = spk;
    }
  }
}

// ---------------- host side ----------------
extern "C" void kernel_launch(void* const* d_in, const int* in_sizes, int n_in,
                              void* d_out, int out_size, void* d_ws, size_t ws_size,
                              hipStream_t stream) {
  (void)in_sizes; (void)n_in; (void)out_size; (void)ws_size;
  const float* inputs = (const float*)d_in[0];
  const float* ann1_w = (const float*)d_in[1];
  const float* ann1_b = (const float*)d_in[2];
  const float* w2 = (const float*)d_in[3];
  const float* b2 = (const float*)d_in[4];
  const float* w3 = (const float*)d_in[5];
  const float* b3 = (const float*)d_in[6];
  const float* w5 = (const float*)d_in[7];
  const float* b5 = (const float*)d_in[8];
  const float* w7 = (const float*)d_in[9];
  const float* b7 = (const float*)d_in[10];
  const float* w8 = (const float*)d_in[11];
  const float* b8 = (const float*)d_in[12];
  float* out = (float*)d_out;

  const size_t FRAME = (size_t)BATCH * 32 * 30 * 30; // 921600
  const size_t S2 = (size_t)BATCH * 32 * 28 * 28;    // 802816
  const size_t S3 = (size_t)BATCH * 64 * 26 * 26;    // 1384448
  const size_t S4 = (size_t)BATCH * 64 * 13 * 13;    // 346112
  const size_t S5 = (size_t)BATCH * 64 * 11 * 11;    // 247808
  const size_t S6 = (size_t)BATCH * 1600;            // 51200
  const size_t S7 = (size_t)BATCH * 512;             // 16384
  const size_t S8 = (size_t)BATCH * 10;              // 320

  float* W = (float*)d_ws;
  size_t off = 0;
  auto alloc = [&](size_t n) { float* p = W + off; off += n; return p; };

  // --- recurrent state (zeroed every launch) ---
  float* ax2y1 = alloc(FRAME); float* ax2y2 = alloc(FRAME);
  float* v2 = alloc(S2); float* s2 = alloc(S2);
  float* ax3y1 = alloc(S2); float* ax3y2 = alloc(S2);
  float* v3 = alloc(S3); float* s3 = alloc(S3);
  float* ax4y1 = alloc(S3); float* ax4y2 = alloc(S3);
  float* ax5y1 = alloc(S4); float* ax5y2 = alloc(S4);
  float* v5 = alloc(S5); float* s5 = alloc(S5);
  float* ax6y1 = alloc(S5); float* ax6y2 = alloc(S5);
  float* ax7y1 = alloc(S6); float* ax7y2 = alloc(S6);
  float* v7 = alloc(S7); float* s7 = alloc(S7);
  float* ax8y1 = alloc(S7); float* ax8y2 = alloc(S7);
  float* v8 = alloc(S8); float* s8 = alloc(S8);
  const size_t stateTotal = off;

  // --- transients (fully overwritten before use) ---
  float* frame = alloc(FRAME);
  float* a2o = alloc(FRAME);
  float* a3o = alloc(S2);
  float* a4o = alloc(S3);
  float* a5o = alloc(S4);
  float* a6o = alloc(S5);
  float* a7o = alloc(S6);
  float* a8o = alloc(S7);

  // zero recurrent state (graph-replay safe, deterministic)
  zero_kernel<<<4096, 256, 0, stream>>>(W, stateTotal);

  // constant ANN front-end
  {
    int n = (int)FRAME;
    ann1_kernel<<<(n + 255) / 256, 256, 0, stream>>>(inputs, ann1_w, ann1_b, frame);
  }

  auto launch_conv = [&](const float* xin, const float* ww, const float* bb,
                         float* v, float* s, float* y1, float* y2, float* ao,
                         int Cin, int Cout, int Hin, int Win, int Hout, int Wout) {
    int M = BATCH * Hout * Wout;
    int tiles = (M / 16) * (Cout / 32);         // one 16x32 tile per wave
    int blocks = (tiles + 3) / 4;               // 4 waves per block (128 threads)
    conv_wmma_lif_axon<<<blocks, 128, 0, stream>>>(xin, ww, bb, v, s, y1, y2, ao,
                                                   Cin, Cout, Hin, Win, Hout, Wout, tiles);
  };

  for (int t = 0; t < TSTEPS; ++t) {
    // axon on constant frame -> a2o
    axon_kernel<<<((int)FRAME + 255) / 256, 256, 0, stream>>>(frame, ax2y1, ax2y2, a2o, (int)FRAME);
    // conv2 (32->32, 30->28) + LIF + axon3
    launch_conv(a2o, w2, b2, v2, s2, ax3y1, ax3y2, a3o, 32, 32, 30, 30, 28, 28);
    // conv3 (32->64, 28->26) + LIF + axon4
    launch_conv(a3o, w3, b3, v3, s3, ax4y1, ax4y2, a4o, 32, 64, 28, 28, 26, 26);
    // maxpool 26->13 + axon5
    {
      int n = (int)S4;
      maxpool_axon_kernel<<<(n + 255) / 256, 256, 0, stream>>>(a4o, ax5y1, ax5y2, a5o,
                                                               64, 26, 26, 13, 13);
    }
    // conv5 (64->64, 13->11) + LIF + axon6
    launch_conv(a5o, w5, b5, v5, s5, ax6y1, ax6y2, a6o, 64, 64, 13, 13, 11, 11);
    // maxpool 11->5 + axon7 (flattened [B,1600])
    {
      int n = (int)S6;
      maxpool_axon_kernel<<<(n + 255) / 256, 256, 0, stream>>>(a6o, ax7y1, ax7y2, a7o,
                                                               64, 11, 11, 5, 5);
    }
    // fc7 (1600->512) + LIF + axon8 : 2 M-tiles x 16 N-pairs = 32 waves
    fc_wmma_lif<<<8, 128, 0, stream>>>(a7o, w7, b7, v7, s7, ax8y1, ax8y2, a8o,
                                       nullptr, 1600, 16, 512, t);
    // fc8 (512->10, masked N-tile) + LIF -> d_out[:, :, t] : 2 waves
    fc_wmma_lif<<<1, 64, 0, stream>>>(a8o, w8, b8, v8, s8, nullptr, nullptr, nullptr,
                                      out, 512, 1, 10, t);
  }
}